// GCNEncoder_54760833024008
// MI455X (gfx1250) — compile-verified
//
#include <hip/hip_runtime.h>

typedef __attribute__((ext_vector_type(2))) float v2f;
typedef __attribute__((ext_vector_type(8))) float v8f;

#define N_NODES 50000
#define N_EDGES 600000
#define IN_CH   128
#define HID     128
#define OUT_CH  64

// ---------------------------------------------------------------- utilities

__global__ void fill_kernel(float* __restrict__ p, float v, int n) {
    int i = blockIdx.x * blockDim.x + threadIdx.x;
    if (i < n) p[i] = v;
}

// deg[dst[e]] += 1 for every real edge (deg pre-initialized to 1.0 = self-loop)
__global__ void degree_kernel(const long long* __restrict__ dst,
                              float* __restrict__ deg, int E) {
    int e = blockIdx.x * blockDim.x + threadIdx.x;
    if (e < E) atomicAdd(&deg[(size_t)dst[e]], 1.0f);
}

// deg -> d^{-1/2} in place
__global__ void rsqrt_kernel(float* __restrict__ deg, int n) {
    int i = blockIdx.x * blockDim.x + threadIdx.x;
    if (i < n) {
        float d = deg[i];
        deg[i] = (d > 0.0f) ? rsqrtf(d) : 0.0f;
    }
}

// ------------------------------------------------- fp32 WMMA GEMM  Out = A*B
// A: [nrows x K] row-major, B: [K x N] row-major, Out: [nrows x N] row-major.
// One wave per 16x16 output tile, V_WMMA_F32_16X16X4_F32, K stepped by 4.
// A layout (ISA 7.12.2, 32-bit A 16x4): lanes 0-15 hold K={k0,k0+1},
// lanes 16-31 hold K={k0+2,k0+3} for row M = lane&15.
// C/D layout: VGPR r -> row r (lanes 0-15) / row r+8 (lanes 16-31).
template <int KDIM, int NDIM>
__global__ __launch_bounds__(32 * (NDIM / 16))
void gemm_wmma_f32(const float* __restrict__ A, const float* __restrict__ B,
                   float* __restrict__ Out) {
    const int lane = threadIdx.x & 31;
    const int wave = threadIdx.x >> 5;
    const int m    = lane & 15;
    const int sel  = lane >> 4;              // 0: K pair {0,1}, 1: K pair {2,3}
    const int row0 = blockIdx.x * 16;        // 50000 % 16 == 0 -> no guards
    const int col0 = wave * 16;

    const float* arow = A + (size_t)(row0 + m) * KDIM + 2 * sel;
    const float* bcol = B + (size_t)(2 * sel) * NDIM + col0 + m;

    v8f acc = {};
#pragma unroll
    for (int k0 = 0; k0 < KDIM; k0 += 4) {
        v2f a;
        a.x = arow[k0];
        a.y = arow[k0 + 1];
        v2f b;
        b.x = bcol[(size_t)k0 * NDIM];
        b.y = bcol[(size_t)(k0 + 1) * NDIM];
        acc = __builtin_amdgcn_wmma_f32_16x16x4_f32(
            /*neg_a=*/false, a, /*neg_b=*/false, b,
            /*c_mod=*/(short)0, acc, /*reuse_a=*/false, /*reuse_b=*/false);
    }

    float* orow = Out + (size_t)(row0 + 8 * sel) * NDIM + col0 + m;
#pragma unroll
    for (int r = 0; r < 8; ++r)
        orow[(size_t)r * NDIM] = acc[r];
}

// --------------------------------------------- edge gather/scale/scatter-add
// 128 channels: one wave per edge, float4 per lane (32*4 = 128).
__global__ __launch_bounds__(256)
void scatter128_kernel(const long long* __restrict__ src,
                       const long long* __restrict__ dst,
                       const float* __restrict__ dinv,
                       const float* __restrict__ H,
                       float* __restrict__ Agg, int E) {
    const int lane = threadIdx.x & 31;
    const int e    = blockIdx.x * 8 + (threadIdx.x >> 5);
    if (e >= E) return;
    const long long s = src[e];
    const long long d = dst[e];
    const float coef = dinv[s] * dinv[d];
    const float4 v = ((const float4*)(H + (size_t)s * 128))[lane];
    float* out = Agg + (size_t)d * 128 + lane * 4;
    atomicAdd(out + 0, v.x * coef);
    atomicAdd(out + 1, v.y * coef);
    atomicAdd(out + 2, v.z * coef);
    atomicAdd(out + 3, v.w * coef);
}

// 64 channels: one wave per edge, float2 per lane (32*2 = 64).
__global__ __launch_bounds__(256)
void scatter64_kernel(const long long* __restrict__ src,
                      const long long* __restrict__ dst,
                      const float* __restrict__ dinv,
                      const float* __restrict__ H,
                      float* __restrict__ Agg, int E) {
    const int lane = threadIdx.x & 31;
    const int e    = blockIdx.x * 8 + (threadIdx.x >> 5);
    if (e >= E) return;
    const long long s = src[e];
    const long long d = dst[e];
    const float coef = dinv[s] * dinv[d];
    const float2 v = ((const float2*)(H + (size_t)s * 64))[lane];
    float* out = Agg + (size_t)d * 64 + lane * 2;
    atomicAdd(out + 0, v.x * coef);
    atomicAdd(out + 1, v.y * coef);
}

// ---------------------------------------------------------------- epilogues
// Agg := relu(Agg + H*dinv^2 + b)  (self-loop term folded in), in place.
__global__ __launch_bounds__(128)
void epilogue_relu128(float* __restrict__ Agg, const float* __restrict__ H,
                      const float* __restrict__ dinv,
                      const float* __restrict__ bias) {
    const int i = blockIdx.x;
    const int c = threadIdx.x;
    const float s = dinv[i];
    const size_t idx = (size_t)i * 128 + c;
    const float v = Agg[idx] + H[idx] * (s * s) + bias[c];
    Agg[idx] = fmaxf(v, 0.0f);
}

// Out := Agg + A1*dinv^2 + b  (final layer, no relu)
__global__ __launch_bounds__(64)
void epilogue_out64(const float* __restrict__ Agg, const float* __restrict__ A1,
                    const float* __restrict__ dinv,
                    const float* __restrict__ bias, float* __restrict__ Out) {
    const int i = blockIdx.x;
    const int c = threadIdx.x;
    const float s = dinv[i];
    const size_t idx = (size_t)i * 64 + c;
    Out[idx] = Agg[idx] + A1[idx] * (s * s) + bias[c];
}

// ---------------------------------------------------------------------- host

static inline size_t align_up(size_t v, size_t a) { return (v + a - 1) / a * a; }

extern "C" void kernel_launch(void* const* d_in, const int* in_sizes, int n_in,
                              void* d_out, int out_size, void* d_ws, size_t ws_size,
                              hipStream_t stream) {
    const float*     x   = (const float*)d_in[0];
    const long long* ei  = (const long long*)d_in[1];   // int64, shape [2, E]
    const float*     W1  = (const float*)d_in[2];
    const float*     b1  = (const float*)d_in[3];
    const float*     W2  = (const float*)d_in[4];
    const float*     b2  = (const float*)d_in[5];
    float*           out = (float*)d_out;

    const long long* src = ei;            // edge_index[0]
    const long long* dst = ei + N_EDGES;  // edge_index[1]

    // workspace layout (~64.3 MB total)
    char* ws = (char*)d_ws;
    size_t off = 0;
    float* deg  = (float*)(ws + off); off = align_up(off + (size_t)N_NODES * 4, 256);
    float* buf1 = (float*)(ws + off); off = align_up(off + (size_t)N_NODES * 128 * 4, 256); // h1, then a1
    float* buf2 = (float*)(ws + off); off = align_up(off + (size_t)N_NODES * 128 * 4, 256); // agg1 -> h2
    float* buf3 = (float*)(ws + off); off = align_up(off + (size_t)N_NODES * 64 * 4, 256);  // agg2
    (void)ws_size; (void)in_sizes; (void)n_in; (void)out_size;

    const int T = 256;
    // ---- normalization: deg = 1 (self-loop) + in-degree; then d^{-1/2}
    fill_kernel<<<(N_NODES + T - 1) / T, T, 0, stream>>>(deg, 1.0f, N_NODES);
    fill_kernel<<<((N_NODES * 128) + T - 1) / T, T, 0, stream>>>(buf2, 0.0f, N_NODES * 128);
    fill_kernel<<<((N_NODES * 64) + T - 1) / T, T, 0, stream>>>(buf3, 0.0f, N_NODES * 64);
    degree_kernel<<<(N_EDGES + T - 1) / T, T, 0, stream>>>(dst, deg, N_EDGES);
    rsqrt_kernel<<<(N_NODES + T - 1) / T, T, 0, stream>>>(deg, N_NODES);

    // ---- layer 1: h1 = x @ W1 (fp32 WMMA), scatter, relu epilogue
    gemm_wmma_f32<IN_CH, HID><<<N_NODES / 16, 32 * (HID / 16), 0, stream>>>(x, W1, buf1);
    scatter128_kernel<<<(N_EDGES + 7) / 8, 256, 0, stream>>>(src, dst, deg, buf1, buf2, N_EDGES);
    epilogue_relu128<<<N_NODES, 128, 0, stream>>>(buf2, buf1, deg, b1);

    // ---- layer 2: a1 = h2 @ W2 (fp32 WMMA), scatter, final epilogue -> d_out
    gemm_wmma_f32<HID, OUT_CH><<<N_NODES / 16, 32 * (OUT_CH / 16), 0, stream>>>(buf2, W2, buf1);
    scatter64_kernel<<<(N_EDGES + 7) / 8, 256, 0, stream>>>(src, dst, deg, buf2, buf3, N_EDGES);
    epilogue_out64<<<N_NODES, 64, 0, stream>>>(buf3, buf1, deg, b2, out);
}